// TinyDLRM_67001489817694
// MI455X (gfx1250) — compile-verified
//
#include <hip/hip_runtime.h>
#include <math.h>

typedef __attribute__((ext_vector_type(2))) float v2f;
typedef __attribute__((ext_vector_type(8))) float v8f;

#define EMBED_DIM 8
#define IN_DIM 26        // 8*3 + 2
#define KPAD 28          // padded to multiple of 4 for 16x16x4 WMMA
#define HIDDEN 16
#define WAVES_PER_BLOCK 8
#define ROWS_PER_WAVE 16
#define ROWS_PER_BLOCK (WAVES_PER_BLOCK * ROWS_PER_WAVE)

// ds_swizzle group-of-32 XOR butterflies (and_mask=0x1f, or=0, xor=1/2/4/8).
// Masks < 16 keep the exchange within each 16-lane half of the wave32.
#define SWZ_ADD(x, imm) \
    (x) += __int_as_float(__builtin_amdgcn_ds_swizzle(__float_as_int(x), (imm)))

__global__ __launch_bounds__(256) void TinyDLRM_wmma_kernel(
    const int* __restrict__ user_id,
    const int* __restrict__ item_id,
    const int* __restrict__ cat_id,
    const float* __restrict__ dense,      // [B,2]
    const float* __restrict__ user_table, // [NU,8]
    const float* __restrict__ item_table, // [NI,8]
    const float* __restrict__ cat_table,  // [NC,8]
    const float* __restrict__ w1,         // [26,16] row-major
    const float* __restrict__ b1,         // [16]
    const float* __restrict__ w2,         // [16]
    const float* __restrict__ b2,         // [1]
    float* __restrict__ out,              // [B]
    int batch)
{
    __shared__ float lds[WAVES_PER_BLOCK * ROWS_PER_WAVE * KPAD];

    const int lane = threadIdx.x & 31;
    const int waveInBlock = threadIdx.x >> 5;
    const int tile = blockIdx.x * WAVES_PER_BLOCK + waveInBlock;
    const int rowBase = tile * ROWS_PER_WAVE;
    if (rowBase >= batch) return;   // wave-uniform

    float* feat = &lds[waveInBlock * ROWS_PER_WAVE * KPAD];

    // ---- Stage 16 rows x 28 features into LDS (no divergence).
    // lane = 2*r + half : each lane builds half of one row's embedding block.
    {
        const int r = lane >> 1;
        const int half = lane & 1;
        const int row = rowBase + r;
        const int uid = user_id[row];
        const int iid = item_id[row];
        const int cid = cat_id[row];
        const float4 u  = *(const float4*)(user_table + (long)uid * EMBED_DIM + half * 4);
        const float4 it = *(const float4*)(item_table + (long)iid * EMBED_DIM + half * 4);
        const float4 ct = *(const float4*)(cat_table  + (long)cid * EMBED_DIM + half * 4);
        float* frow = feat + r * KPAD;
        *(float4*)(frow + 0  + half * 4) = u;
        *(float4*)(frow + 8  + half * 4) = it;
        *(float4*)(frow + 16 + half * 4) = ct;
        // dense slice of this tile = 32 contiguous floats = 1 per lane (coalesced)
        frow[24 + half] = dense[(long)rowBase * 2 + lane];
        frow[26 + half] = 0.0f;   // K padding
    }
    __syncthreads();

    // ---- WMMA fragment indexing (V_WMMA_F32_16X16X4_F32, wave32):
    //  A vGPR pair: lane holds M = lane%16, K = kb*4 + 2*(lane/16) + {0,1}
    //  B vGPR pair: lane holds N = lane%16, same K mapping
    //  C/D: vgpr v <-> row v + 8*(lane/16), col lane%16
    const int col = lane & 15;
    const int khi = (lane >> 4) << 1;   // 0 or 2

    // Preload B fragments from w1 (tiny, L2/L1 resident), zero-pad K>=26.
    v2f bfrag[7];
#pragma unroll
    for (int kb = 0; kb < 7; ++kb) {
        const int kk = kb * 4 + khi;
        bfrag[kb].x = (kk     < IN_DIM) ? w1[kk * HIDDEN + col]       : 0.0f;
        bfrag[kb].y = (kk + 1 < IN_DIM) ? w1[(kk + 1) * HIDDEN + col] : 0.0f;
    }

    v8f c = {0.f, 0.f, 0.f, 0.f, 0.f, 0.f, 0.f, 0.f};
    const float* frow = feat + col * KPAD;   // A row for this lane
#pragma unroll
    for (int kb = 0; kb < 7; ++kb) {
        const int kk = kb * 4 + khi;
        v2f a;
        a.x = frow[kk];
        a.y = frow[kk + 1];
        // 8 args: (neg_a, A, neg_b, B, c_mod, C, reuse_a, reuse_b)
        c = __builtin_amdgcn_wmma_f32_16x16x4_f32(
                false, a, false, bfrag[kb], (short)0, c, false, false);
    }

    // ---- Layer 2: relu, dot with w2 across the 16 columns, bias2, sigmoid.
    const float bias1 = b1[col];
    const float w2v   = w2[col];
    const float bias2 = b2[0];

    float p[8];
#pragma unroll
    for (int v = 0; v < 8; ++v) {
        float h = c[v] + bias1;
        h = h > 0.0f ? h : 0.0f;
        p[v] = h * w2v;
    }
    // Reduce across the 16 lanes of each half-wave via ds_swizzle XOR butterflies.
#pragma unroll
    for (int v = 0; v < 8; ++v) {
        SWZ_ADD(p[v], 0x041F);   // xor 1
        SWZ_ADD(p[v], 0x081F);   // xor 2
        SWZ_ADD(p[v], 0x101F);   // xor 4
        SWZ_ADD(p[v], 0x201F);   // xor 8
    }

    // lane 0 holds rows 0..7, lane 16 holds rows 8..15 of this tile.
    if (col == 0) {
        float s[8];
#pragma unroll
        for (int v = 0; v < 8; ++v) {
            const float x = p[v] + bias2;
            s[v] = 1.0f / (1.0f + expf(-x));
        }
        float* op = out + rowBase + ((lane >> 4) << 3);
        float4 lo = {s[0], s[1], s[2], s[3]};
        float4 hi = {s[4], s[5], s[6], s[7]};
        *(float4*)(op + 0) = lo;
        *(float4*)(op + 4) = hi;
    }
}

extern "C" void kernel_launch(void* const* d_in, const int* in_sizes, int n_in,
                              void* d_out, int out_size, void* d_ws, size_t ws_size,
                              hipStream_t stream) {
    const int*   user_id    = (const int*)  d_in[0];
    const int*   item_id    = (const int*)  d_in[1];
    const int*   cat_id     = (const int*)  d_in[2];
    const float* dense      = (const float*)d_in[3];
    const float* user_table = (const float*)d_in[4];
    const float* item_table = (const float*)d_in[5];
    const float* cat_table  = (const float*)d_in[6];
    const float* w1         = (const float*)d_in[7];
    const float* b1         = (const float*)d_in[8];
    const float* w2         = (const float*)d_in[9];
    const float* b2         = (const float*)d_in[10];
    float* out = (float*)d_out;

    const int batch = in_sizes[0];
    const int blocks = (batch + ROWS_PER_BLOCK - 1) / ROWS_PER_BLOCK;

    TinyDLRM_wmma_kernel<<<blocks, 256, 0, stream>>>(
        user_id, item_id, cat_id, dense,
        user_table, item_table, cat_table,
        w1, b1, w2, b2, out, batch);
}